// STDPLinear_65652870086857
// MI455X (gfx1250) — compile-verified
//
#include <hip/hip_runtime.h>

// ---------------------------------------------------------------------------
// STDP Linear layer for MI455X (gfx1250, wave32, WMMA + Tensor Data Mover).
//   GEMM1: weighted = in_spikes @ weight.T  (f32 WMMA 16x16x4, K=4096)
//          staged into LDS by TDM (tensor_load_to_lds), double-buffered,
//          overlapped with WMMA compute, synced with s_wait_tensorcnt.
//   LIF:   mem/spikes/tq fused into GEMM1 epilogue
//   tp:    elementwise
//   GEMM2: w_new = clip(weight + 0.001 * tq^T @ tp, -0.1, 0.5), K=64
// Memory-bound problem (~36 MB HBM traffic => ~1.5 us floor at 23.3 TB/s);
// FP32 WMMA keeps reference numerics while matrix units are not the bottleneck.
// ---------------------------------------------------------------------------

typedef float v2f __attribute__((ext_vector_type(2)));
typedef float v8f __attribute__((ext_vector_type(8)));
typedef unsigned int v4u __attribute__((ext_vector_type(4)));
typedef int v4i __attribute__((ext_vector_type(4)));
typedef int v8i __attribute__((ext_vector_type(8)));

#define SNN_B    64
#define SNN_IN   4096
#define SNN_OUT  1024
#define SNN_BETA 0.98f
#define SNN_THRESH 0.4f
#define SNN_DECAY 0.85f
#define SNN_LR (0.005f - 0.004f)

#define KC 64              // K-chunk for kernel 1
#define LROW (KC + 2)      // padded LDS row stride (floats) -> no bank conflicts

__device__ __forceinline__ float clip01(float v) {
    return fminf(fmaxf(v, 0.0f), 1.0f);
}

// ---------------------------------------------------------------------------
// Tensor Data Mover: 2D f32 tile load Global -> LDS with padded rows.
// D# built per CDNA5 ISA 8.3-8.6:
//   group0: count=1 | lds_addr | global_addr(57b) | type=2
//   group1: data_size=4B, pad_enable=1, pad_interval=5 (64 DWORDs),
//           pad_amount=1 (2 DWORDs) -> LDS row stride 66 floats,
//           tensor dims/strides + tile dims; groups 2/3 zero (2D tensor).
// ---------------------------------------------------------------------------
__device__ __forceinline__ void tdm_load_2d(unsigned lds_byte_addr,
                                            const void* gptr,
                                            unsigned tensor_d0,   // row len (elems)
                                            unsigned tensor_d1,   // #rows
                                            unsigned tile_d0,     // tile row len
                                            unsigned tile_d1,     // tile #rows
                                            unsigned stride_d0)   // row stride (elems)
{
    unsigned long long ga = (unsigned long long)(uintptr_t)gptr;
    v4u g0;
    g0[0] = 1u;                                            // count=1, load, no gather
    g0[1] = lds_byte_addr;
    g0[2] = (unsigned)(ga & 0xFFFFFFFFu);
    g0[3] = (unsigned)((ga >> 32) & 0x01FFFFFFu) | 0x80000000u;  // type=2 (image)

    v8i g1;
    g1[0] = (int)((2u << 16) |          // data_size = 4B
                  (1u << 20) |          // pad_enable
                  (5u << 22) |          // pad_interval: 64 DWORDs
                  (1u << 25));          // pad_amount: 2 DWORDs
    g1[1] = (int)((tensor_d0 & 0xFFFFu) << 16);                     // abar=0 | td0.lo
    g1[2] = (int)((tensor_d0 >> 16) | ((tensor_d1 & 0xFFFFu) << 16));
    g1[3] = (int)((tensor_d1 >> 16) | ((tile_d0 & 0xFFFFu) << 16));
    g1[4] = (int)(tile_d1 & 0xFFFFu);                               // tile_dim2 = 0
    g1[5] = (int)stride_d0;                                         // stride0 lo32
    g1[6] = 0;                                                      // stride0 hi / stride1 lo
    g1[7] = 0;

    v4i z4 = {0, 0, 0, 0};
#if defined(__clang_major__) && (__clang_major__ >= 23)
    v8i z8 = {0, 0, 0, 0, 0, 0, 0, 0};
    __builtin_amdgcn_tensor_load_to_lds(g0, g1, z4, z4, z8, 0);
#else
    __builtin_amdgcn_tensor_load_to_lds(g0, g1, z4, z4, 0);
#endif
}

// ---------------------------------------------------------------------------
// Kernel 1: weighted = x @ w^T  ->  LIF update  ->  spikes, mem, tq
// Block: 256 threads (8 waves). Block tile: M=64 (batch) x N=32 outs.
// TDM double-buffers the A/W LDS tiles; wave 0 drives the DMA, all 8 waves
// run v_wmma_f32_16x16x4_f32 (16 k-steps per chunk, 64 chunks).
// ---------------------------------------------------------------------------
__global__ __launch_bounds__(256)
void snn_lif_gemm(const float* __restrict__ x,          // [B,IN]
                  const float* __restrict__ w,          // [OUT,IN]
                  const float* __restrict__ membrane,   // [B,OUT]
                  const float* __restrict__ trace_post, // [OUT]
                  float* __restrict__ out_spikes,       // [B,OUT]
                  float* __restrict__ out_mem,          // [B,OUT]
                  float* __restrict__ out_tq)           // [B,OUT]
{
    __shared__ float A_lds[2][SNN_B * LROW];   // 2 x 64 x 66 floats
    __shared__ float W_lds[2][32 * LROW];      // 2 x 32 x 66 floats

    const int tid   = threadIdx.x;
    const int wave  = tid >> 5;             // wave32
    const int lane  = tid & 31;
    const int obase = blockIdx.x * 32;
    const int mtile = (wave & 3) * 16;      // batch-row tile: 0/16/32/48
    const int ntile = (wave >> 2) * 16;     // out-col tile within block: 0/16

    const int lmod  = lane & 15;
    const int lhalf = (lane < 16) ? 0 : 2;  // f32 WMMA A/B half-wave K split

    v8f acc = {0.f, 0.f, 0.f, 0.f, 0.f, 0.f, 0.f, 0.f};

    // prologue: TDM chunk 0 into buffer 0 (wave 0 drives the DMA engine)
    if (wave == 0) {
        tdm_load_2d((unsigned)(uintptr_t)&A_lds[0][0], &x[0],
                    SNN_IN, SNN_B, KC, SNN_B, SNN_IN);
        tdm_load_2d((unsigned)(uintptr_t)&W_lds[0][0], &w[(size_t)obase * SNN_IN],
                    SNN_IN, SNN_OUT, KC, 32, SNN_IN);
    }

    int cur = 0;
    for (int kc = 0; kc < SNN_IN; kc += KC) {
        if (wave == 0) {
            if (kc + KC < SNN_IN) {
                // issue next chunk into the other buffer, then wait for the
                // current chunk (leave the 2 newest TDM ops outstanding)
                tdm_load_2d((unsigned)(uintptr_t)&A_lds[cur ^ 1][0],
                            &x[kc + KC], SNN_IN, SNN_B, KC, SNN_B, SNN_IN);
                tdm_load_2d((unsigned)(uintptr_t)&W_lds[cur ^ 1][0],
                            &w[(size_t)obase * SNN_IN + kc + KC],
                            SNN_IN, SNN_OUT, KC, 32, SNN_IN);
                __builtin_amdgcn_s_wait_tensorcnt(2);
            } else {
                __builtin_amdgcn_s_wait_tensorcnt(0);
            }
        }
        __syncthreads();   // publish buffer `cur` to all waves

        const float* arow = &A_lds[cur][(mtile + lmod) * LROW + lhalf];
        const float* brow = &W_lds[cur][(ntile + lmod) * LROW + lhalf];
        #pragma unroll
        for (int ks = 0; ks < KC; ks += 4) {
            v2f a = *reinterpret_cast<const v2f*>(arow + ks);
            v2f b = *reinterpret_cast<const v2f*>(brow + ks);
            acc = __builtin_amdgcn_wmma_f32_16x16x4_f32(
                false, a, false, b, (short)0, acc, false, false);
        }
        __syncthreads();   // all waves done with buffer `cur` before reuse
        cur ^= 1;
    }

    // ---- LIF epilogue: mem = mem*beta + weighted; spike; reset; tq ----
    const int ocol  = obase + ntile + lmod;
    const int rbase = mtile + ((lane >> 4) << 3);   // +8 for lanes 16..31
    const float tpost = trace_post[ocol] * SNN_DECAY;
    #pragma unroll
    for (int r = 0; r < 8; ++r) {
        const int m   = rbase + r;
        const int idx = m * SNN_OUT + ocol;
        float mem   = membrane[idx] * SNN_BETA + acc[r];
        float spike = (mem > SNN_THRESH) ? 1.0f : 0.0f;
        mem = (spike > 0.0f) ? 0.0f : mem;
        out_spikes[idx] = spike;
        out_mem[idx]    = mem;
        out_tq[idx]     = clip01(tpost + spike);
    }
}

// ---------------------------------------------------------------------------
// Kernel 2: tp = clip(trace_pre * decay + in_spikes, 0, 1)   [B,IN]
// ---------------------------------------------------------------------------
__global__ __launch_bounds__(256)
void snn_tp(const float* __restrict__ x,          // [B,IN]
            const float* __restrict__ trace_pre,  // [IN]
            float* __restrict__ out_tp)           // [B,IN]
{
    const int idx = (blockIdx.x * 256 + threadIdx.x) << 2;   // float4 granular
    const int i   = idx & (SNN_IN - 1);
    float4 xs = *reinterpret_cast<const float4*>(&x[idx]);
    float4 tr = *reinterpret_cast<const float4*>(&trace_pre[i]);
    float4 o;
    o.x = clip01(tr.x * SNN_DECAY + xs.x);
    o.y = clip01(tr.y * SNN_DECAY + xs.y);
    o.z = clip01(tr.z * SNN_DECAY + xs.z);
    o.w = clip01(tr.w * SNN_DECAY + xs.w);
    *reinterpret_cast<float4*>(&out_tp[idx]) = o;
}

// ---------------------------------------------------------------------------
// Kernel 3: w_new = clip(w + lr * (tq^T @ tp), -0.1, 0.5)
// A = tq^T [OUT x 64], B = tp [64 x IN], K = batch = 64 (single LDS stage).
// Block tile: 64 (out) x 64 (in); 8 waves, 2 accumulators each.
// ---------------------------------------------------------------------------
#define ST3 (64 + 2)

__global__ __launch_bounds__(256)
void snn_stdp_gemm(const float* __restrict__ tp,   // [B,IN]
                   const float* __restrict__ tq,   // [B,OUT]
                   const float* __restrict__ w,    // [OUT,IN]
                   float* __restrict__ w_new)      // [OUT,IN]
{
    __shared__ float Atq[64 * ST3];   // [o_local][b]  (tq transposed)
    __shared__ float Btp[64 * ST3];   // [i_local][b]  (tp transposed)

    const int tid   = threadIdx.x;
    const int wave  = tid >> 5;
    const int lane  = tid & 31;
    const int obase = (blockIdx.x & 15) * 64;   // OUT/64 = 16
    const int ibase = (blockIdx.x >> 4) * 64;   // IN/64  = 64

    // prefetch the weight tile needed by the epilogue (global_prefetch_b8)
    {
        int prow = tid >> 2;                    // 64 rows, 4 threads each
        int pcol = (tid & 3) << 4;              // 16-float stride
        __builtin_prefetch(&w[(size_t)(obase + prow) * SNN_IN + ibase + pcol], 0, 1);
    }

    // stage + transpose: coalesced global reads, strided LDS writes (tiny)
    #pragma unroll
    for (int q = 0; q < 16; ++q) {
        int j = tid + 256 * q;                  // 0..4095
        int b = j >> 6;
        int c = j & 63;
        Atq[c * ST3 + b] = tq[b * SNN_OUT + obase + c];
        Btp[c * ST3 + b] = tp[b * SNN_IN  + ibase + c];
    }
    __syncthreads();

    const int lmod  = lane & 15;
    const int lhalf = (lane < 16) ? 0 : 2;
    const int mtile = (wave & 3) * 16;          // out tile
    const int nt0   = (wave >> 2) * 32;         // two in tiles: nt0, nt0+16

    v8f acc0 = {0.f, 0.f, 0.f, 0.f, 0.f, 0.f, 0.f, 0.f};
    v8f acc1 = {0.f, 0.f, 0.f, 0.f, 0.f, 0.f, 0.f, 0.f};

    const float* arow  = &Atq[(mtile + lmod) * ST3 + lhalf];
    const float* brow0 = &Btp[(nt0 + lmod) * ST3 + lhalf];
    const float* brow1 = &Btp[(nt0 + 16 + lmod) * ST3 + lhalf];

    #pragma unroll
    for (int ks = 0; ks < 64; ks += 4) {        // K = batch = 64
        v2f a  = *reinterpret_cast<const v2f*>(arow + ks);
        v2f b0 = *reinterpret_cast<const v2f*>(brow0 + ks);
        v2f b1 = *reinterpret_cast<const v2f*>(brow1 + ks);
        acc0 = __builtin_amdgcn_wmma_f32_16x16x4_f32(
            false, a, false, b0, (short)0, acc0, false, false);
        acc1 = __builtin_amdgcn_wmma_f32_16x16x4_f32(
            false, a, false, b1, (short)0, acc1, false, false);
    }

    // epilogue: read weight, apply dw, clip, write w_new
    const int rbase = mtile + ((lane >> 4) << 3);
    #pragma unroll
    for (int r = 0; r < 8; ++r) {
        const int o  = obase + rbase + r;
        const int i0 = ibase + nt0 + lmod;
        const long base = (long)o * SNN_IN;
        float v0 = w[base + i0]      + SNN_LR * acc0[r];
        float v1 = w[base + i0 + 16] + SNN_LR * acc1[r];
        w_new[base + i0]      = fminf(fmaxf(v0, -0.1f), 0.5f);
        w_new[base + i0 + 16] = fminf(fmaxf(v1, -0.1f), 0.5f);
    }
}

// ---------------------------------------------------------------------------
extern "C" void kernel_launch(void* const* d_in, const int* in_sizes, int n_in,
                              void* d_out, int out_size, void* d_ws, size_t ws_size,
                              hipStream_t stream) {
    (void)in_sizes; (void)n_in; (void)out_size; (void)d_ws; (void)ws_size;

    const float* x          = (const float*)d_in[0];  // in_spikes [64,4096]
    const float* w          = (const float*)d_in[1];  // weight    [1024,4096]
    const float* membrane   = (const float*)d_in[2];  // [64,1024]
    const float* trace_pre  = (const float*)d_in[3];  // [4096]
    const float* trace_post = (const float*)d_in[4];  // [1024]

    float* out        = (float*)d_out;                // concat: spikes,mem,w_new,tp,tq
    float* out_spikes = out;
    float* out_mem    = out_spikes + SNN_B * SNN_OUT;
    float* out_wnew   = out_mem    + SNN_B * SNN_OUT;
    float* out_tp     = out_wnew   + SNN_OUT * SNN_IN;
    float* out_tq     = out_tp     + SNN_B * SNN_IN;

    // GEMM1 + LIF + tq : 32 blocks x 256 threads
    snn_lif_gemm<<<SNN_OUT / 32, 256, 0, stream>>>(
        x, w, membrane, trace_post, out_spikes, out_mem, out_tq);

    // tp elementwise : 64*4096/4/256 = 256 blocks
    snn_tp<<<(SNN_B * SNN_IN / 4) / 256, 256, 0, stream>>>(x, trace_pre, out_tp);

    // GEMM2 + weight clip : (1024/64)*(4096/64) = 1024 blocks
    snn_stdp_gemm<<<(SNN_OUT / 64) * (SNN_IN / 64), 256, 0, stream>>>(
        out_tp, out_tq, w, out_wnew);
}